// Mamba_31190052503567
// MI455X (gfx1250) — compile-verified
//
#include <hip/hip_runtime.h>
#include <hip/hip_bf16.h>

// ---------------- model constants ----------------
#define LSEQ     1024
#define DMODEL   1024
#define NLAYERS  8
#define DINNER   2048
#define HEADDIM  64
#define NHEADS   32
#define DSTATE   128
#define DCONV    4
#define CHUNK    256
#define NCHUNK   (LSEQ / CHUNK)
#define CONVDIM  (DINNER + 2 * DSTATE)          // 2304
#define DINPROJ  (2 * DINNER + 2 * DSTATE + NHEADS) // 4384
#define VOCAB    32000
#define EPS      1e-5f

typedef float v2f __attribute__((ext_vector_type(2)));
typedef float v8f __attribute__((ext_vector_type(8)));

// ======================================================================
// Generic fp32 WMMA GEMM:  C[M,N] = A[M,K] * B[N,K]^T   (both row-major,
// K contiguous -> both operands load coalesced). M % 128 == 0, K % 16 == 0,
// N ragged (guarded). Block = 256 threads = 8 waves; block tile 128x128x16;
// wave tile 64x32 = 4x2 WMMA 16x16 tiles. LDS double-buffered: global
// prefetch of tile k+1 overlaps the v_wmma burst on tile k; one barrier
// per k-tile.
// ======================================================================
__global__ __launch_bounds__(256) void gemm_tn_wmma(
    const float* __restrict__ A, const float* __restrict__ B,
    float* __restrict__ C, int M, int N, int K)
{
  constexpr int BM = 128, BN = 128, BK = 16, LDT = BK + 1; // pad -> no bank conflicts
  __shared__ float As[2][BM * LDT];
  __shared__ float Bs[2][BN * LDT];

  const int tid  = threadIdx.x;
  const int m0   = blockIdx.y * BM;
  const int n0   = blockIdx.x * BN;
  const int wid  = tid >> 5;
  const int lane = tid & 31;
  const int wm   = wid & 1;         // 2 waves along M (64 rows each)
  const int wn   = wid >> 1;        // 4 waves along N (32 cols each)
  const int lm   = lane & 15;       // fragment row/col within 16
  const int lg   = lane >> 4;       // half-wave group

  v8f acc[4][2];
  #pragma unroll
  for (int i = 0; i < 4; ++i)
    #pragma unroll
    for (int j = 0; j < 2; ++j)
      #pragma unroll
      for (int e = 0; e < 8; ++e) acc[i][j][e] = 0.f;

  const int lr = tid >> 1;          // 0..127 tile row
  const int lc = (tid & 1) * 8;     // 0 or 8
  const int nr = n0 + lr;           // B row (ragged guarded)
  const float* gaRow = A + (size_t)(m0 + lr) * K + lc;
  const float* gbRow = B + (size_t)nr * K + lc;

  float ra[8], rb[8];

  // ---- prologue: stage k-tile 0 ----
  #pragma unroll
  for (int i = 0; i < 8; ++i) ra[i] = gaRow[i];
  #pragma unroll
  for (int i = 0; i < 8; ++i) rb[i] = (nr < N) ? gbRow[i] : 0.f;
  #pragma unroll
  for (int i = 0; i < 8; ++i) As[0][lr * LDT + lc + i] = ra[i];
  #pragma unroll
  for (int i = 0; i < 8; ++i) Bs[0][lr * LDT + lc + i] = rb[i];
  __syncthreads();

  const int ktiles = K / BK;
  int cur = 0;
  for (int kt = 0; kt < ktiles; ++kt) {
    // ---- prefetch k-tile kt+1 into registers (overlaps compute) ----
    if (kt + 1 < ktiles) {
      const float* ga = gaRow + (size_t)(kt + 1) * BK;
      #pragma unroll
      for (int i = 0; i < 8; ++i) ra[i] = ga[i];
      const float* gb = gbRow + (size_t)(kt + 1) * BK;
      #pragma unroll
      for (int i = 0; i < 8; ++i) rb[i] = (nr < N) ? gb[i] : 0.f;
    }

    // ---- compute on current LDS buffer ----
    const float* Ab = As[cur];
    const float* Bb = Bs[cur];
    #pragma unroll
    for (int kk = 0; kk < BK; kk += 4) {
      v2f bfrag[2];
      #pragma unroll
      for (int j = 0; j < 2; ++j) {
        const int ncol = wn * 32 + j * 16 + lm;
        bfrag[j].x = Bb[ncol * LDT + kk + lg * 2 + 0];
        bfrag[j].y = Bb[ncol * LDT + kk + lg * 2 + 1];
      }
      #pragma unroll
      for (int i = 0; i < 4; ++i) {
        const int mrow = wm * 64 + i * 16 + lm;
        v2f afrag;
        afrag.x = Ab[mrow * LDT + kk + lg * 2 + 0];
        afrag.y = Ab[mrow * LDT + kk + lg * 2 + 1];
        #pragma unroll
        for (int j = 0; j < 2; ++j) {
          acc[i][j] = __builtin_amdgcn_wmma_f32_16x16x4_f32(
              false, afrag, false, bfrag[j], (short)0, acc[i][j], false, false);
        }
      }
    }

    // ---- commit prefetched tile to the alternate LDS buffer ----
    if (kt + 1 < ktiles) {
      const int nxt = cur ^ 1;
      #pragma unroll
      for (int i = 0; i < 8; ++i) As[nxt][lr * LDT + lc + i] = ra[i];
      #pragma unroll
      for (int i = 0; i < 8; ++i) Bs[nxt][lr * LDT + lc + i] = rb[i];
    }
    __syncthreads();
    cur ^= 1;
  }

  // Store: C/D layout -> lane n = lane&15, VGPR v holds m = 8*(lane>>4)+v
  #pragma unroll
  for (int i = 0; i < 4; ++i) {
    const int mrow = m0 + wm * 64 + i * 16 + 8 * lg;
    #pragma unroll
    for (int j = 0; j < 2; ++j) {
      const int ncol = n0 + wn * 32 + j * 16 + lm;
      if (ncol < N) {
        #pragma unroll
        for (int v = 0; v < 8; ++v)
          C[(size_t)(mrow + v) * N + ncol] = acc[i][j][v];
      }
    }
  }
}

// ======================================================================
// Embedding gather + residual zero-init
// ======================================================================
__global__ __launch_bounds__(256) void embed_gather(
    const int* __restrict__ ids, const float* __restrict__ emb,
    float* __restrict__ h, float* __restrict__ res)
{
  const int idx = blockIdx.x * 256 + threadIdx.x;
  if (idx >= LSEQ * DMODEL) return;
  const int t = idx >> 10, d = idx & (DMODEL - 1);
  h[idx]   = emb[(size_t)ids[t] * DMODEL + d];
  res[idx] = 0.f;
}

// ======================================================================
// res = h + res;  out = rms(res) * w   (D = 1024, block per row)
// ======================================================================
__global__ __launch_bounds__(256) void rms_residual(
    const float* __restrict__ h, float* __restrict__ res,
    float* __restrict__ out, const float* __restrict__ w)
{
  const int row = blockIdx.x, tid = threadIdx.x;
  __shared__ float red[8];
  __shared__ float stot;
  float vv[4];
  float ss = 0.f;
  #pragma unroll
  for (int k = 0; k < 4; ++k) {
    const int i = k * 256 + tid;
    float v = h[(size_t)row * DMODEL + i] + res[(size_t)row * DMODEL + i];
    res[(size_t)row * DMODEL + i] = v;
    vv[k] = v;
    ss += v * v;
  }
  for (int o = 16; o > 0; o >>= 1) ss += __shfl_xor(ss, o, 32);
  if ((tid & 31) == 0) red[tid >> 5] = ss;
  __syncthreads();
  if (tid == 0) {
    float t = 0.f;
    #pragma unroll
    for (int i = 0; i < 8; ++i) t += red[i];
    stot = rsqrtf(t / (float)DMODEL + EPS);
  }
  __syncthreads();
  const float sc = stot;
  #pragma unroll
  for (int k = 0; k < 4; ++k) {
    const int i = k * 256 + tid;
    out[(size_t)row * DMODEL + i] = vv[k] * sc * w[i];
  }
}

// ======================================================================
// dt = softplus(zx[:, 4352:4384] + dt_bias)
// ======================================================================
__global__ __launch_bounds__(256) void dt_softplus(
    const float* __restrict__ zx, const float* __restrict__ dt_bias,
    float* __restrict__ dtb)
{
  const int idx = blockIdx.x * 256 + threadIdx.x;
  if (idx >= LSEQ * NHEADS) return;
  const int t = idx >> 5, hh = idx & 31;
  const float x = zx[(size_t)t * DINPROJ + (DINNER + CONVDIM) + hh] + dt_bias[hh];
  dtb[idx] = (x > 20.f) ? x : log1pf(expf(x));
}

// ======================================================================
// Depthwise causal conv (width 4) + bias + SiLU over xBC slice of zx
// ======================================================================
__global__ __launch_bounds__(256) void conv_silu(
    const float* __restrict__ zx, const float* __restrict__ cw,
    const float* __restrict__ cb, float* __restrict__ xbc)
{
  const int idx = blockIdx.x * 256 + threadIdx.x;
  if (idx >= LSEQ * CONVDIM) return;
  const int t = idx / CONVDIM, ch = idx % CONVDIM;
  float acc = cb[ch];
  #pragma unroll
  for (int k = 0; k < DCONV; ++k) {
    const int tt = t - (DCONV - 1) + k;
    if (tt >= 0)
      acc += zx[(size_t)tt * DINPROJ + DINNER + ch] * cw[ch * DCONV + k];
  }
  xbc[idx] = acc / (1.f + expf(-acc));
}

// ======================================================================
// SSD phase A: per (chunk, head) in-chunk sequential scan with zero init.
// Thread (p = tid>>2, q = tid&3) owns 32 state columns n in [32q,32q+32).
// Produces local y (incl. D skip), chunk-final states, inclusive cum-decay.
// ======================================================================
__global__ __launch_bounds__(256) void ssd_local(
    const float* __restrict__ xbc,   // [L, CONVDIM]
    const float* __restrict__ dtb,   // [L, NHEADS]
    const float* __restrict__ A_log, // [NHEADS] layer slice
    const float* __restrict__ Dskip, // [NHEADS]
    float* __restrict__ y,           // [L, DINNER]
    float* __restrict__ states,      // [NCHUNK, NHEADS, HEADDIM, DSTATE]
    float* __restrict__ cumdec)      // [NCHUNK, NHEADS, CHUNK]
{
  const int c = blockIdx.x >> 5;
  const int h = blockIdx.x & 31;
  const int tid = threadIdx.x;
  const int p = tid >> 2, q = tid & 3;
  const int nb = 32 * q;
  const float Ah = -expf(A_log[h]);
  const float Dv = Dskip[h];

  __shared__ float bc[2 * DSTATE]; // B row then C row
  __shared__ float xs[HEADDIM];

  float s[32];
  #pragma unroll
  for (int j = 0; j < 32; ++j) s[j] = 0.f;
  float r = 1.f;

  for (int tin = 0; tin < CHUNK; ++tin) {
    const int t = c * CHUNK + tin;
    bc[tid] = xbc[(size_t)t * CONVDIM + DINNER + tid];
    if (tid < HEADDIM) xs[tid] = xbc[(size_t)t * CONVDIM + h * HEADDIM + tid];
    __syncthreads();

    const float dtv   = dtb[t * NHEADS + h];
    const float decay = expf(dtv * Ah);
    r *= decay;
    if (tid == 0) cumdec[(c * NHEADS + h) * CHUNK + tin] = r;

    const float xv  = xs[p];
    const float dtx = dtv * xv;
    float acc = 0.f;
    #pragma unroll
    for (int j = 0; j < 32; ++j) {
      s[j] = s[j] * decay + dtx * bc[nb + j];
      acc += s[j] * bc[DSTATE + nb + j];
    }
    acc += __shfl_xor(acc, 1, 32);
    acc += __shfl_xor(acc, 2, 32);
    if (q == 0) y[(size_t)t * DINNER + h * HEADDIM + p] = acc + Dv * xv;
    __syncthreads();
  }

  #pragma unroll
  for (int j = 0; j < 32; ++j)
    states[((size_t)(c * NHEADS + h) * HEADDIM + p) * DSTATE + nb + j] = s[j];
}

// ======================================================================
// SSD phase B: tiny cross-chunk scan. prev[c] = state entering chunk c.
// ======================================================================
__global__ __launch_bounds__(256) void ssd_scan(
    const float* __restrict__ states, const float* __restrict__ cumdec,
    float* __restrict__ prev)
{
  const int h = blockIdx.x;
  const int tid = threadIdx.x;
  float P[32];
  #pragma unroll
  for (int j = 0; j < 32; ++j) P[j] = 0.f;
  for (int c = 0; c < NCHUNK; ++c) {
    const size_t base = (size_t)(c * NHEADS + h) * (HEADDIM * DSTATE);
    #pragma unroll
    for (int j = 0; j < 32; ++j) prev[base + j * 256 + tid] = P[j];
    const float ad = cumdec[(c * NHEADS + h) * CHUNK + (CHUNK - 1)];
    #pragma unroll
    for (int j = 0; j < 32; ++j)
      P[j] = ad * P[j] + states[base + j * 256 + tid];
  }
}

// ======================================================================
// SSD phase C: y[t] += cumdec[t] * (C[t] . prev_state), chunk-parallel.
// Same thread layout as phase A; prev held in registers.
// ======================================================================
__global__ __launch_bounds__(256) void ssd_cross(
    const float* __restrict__ xbc, const float* __restrict__ prev,
    const float* __restrict__ cumdec, float* __restrict__ y)
{
  const int c = blockIdx.x >> 5;
  const int h = blockIdx.x & 31;
  const int tid = threadIdx.x;
  const int p = tid >> 2, q = tid & 3;
  const int nb = 32 * q;
  __shared__ float cs[DSTATE];

  float P[32];
  const size_t base = (size_t)(c * NHEADS + h) * (HEADDIM * DSTATE) + p * DSTATE + nb;
  #pragma unroll
  for (int j = 0; j < 32; ++j) P[j] = prev[base + j];

  for (int tin = 0; tin < CHUNK; ++tin) {
    const int t = c * CHUNK + tin;
    if (tid < DSTATE) cs[tid] = xbc[(size_t)t * CONVDIM + DINNER + DSTATE + tid];
    __syncthreads();
    float acc = 0.f;
    #pragma unroll
    for (int j = 0; j < 32; ++j) acc += P[j] * cs[nb + j];
    acc += __shfl_xor(acc, 1, 32);
    acc += __shfl_xor(acc, 2, 32);
    if (q == 0) {
      const float a = cumdec[(c * NHEADS + h) * CHUNK + tin];
      y[(size_t)t * DINNER + h * HEADDIM + p] += a * acc;
    }
    __syncthreads();
  }
}

// ======================================================================
// gated = rms(y * silu(z)) * gate_w   (D = 2048, block per row)
// ======================================================================
__global__ __launch_bounds__(256) void gate_rms(
    const float* __restrict__ y, const float* __restrict__ zx,
    const float* __restrict__ gw, float* __restrict__ out)
{
  const int row = blockIdx.x, tid = threadIdx.x;
  __shared__ float red[8];
  __shared__ float stot;
  float vv[8];
  float ss = 0.f;
  #pragma unroll
  for (int k = 0; k < 8; ++k) {
    const int i = k * 256 + tid;
    const float z = zx[(size_t)row * DINPROJ + i];
    const float v = y[(size_t)row * DINNER + i] * (z / (1.f + expf(-z)));
    vv[k] = v;
    ss += v * v;
  }
  for (int o = 16; o > 0; o >>= 1) ss += __shfl_xor(ss, o, 32);
  if ((tid & 31) == 0) red[tid >> 5] = ss;
  __syncthreads();
  if (tid == 0) {
    float t = 0.f;
    #pragma unroll
    for (int i = 0; i < 8; ++i) t += red[i];
    stot = rsqrtf(t / (float)DINNER + EPS);
  }
  __syncthreads();
  const float sc = stot;
  #pragma unroll
  for (int k = 0; k < 8; ++k) {
    const int i = k * 256 + tid;
    out[(size_t)row * DINNER + i] = vv[k] * sc * gw[i];
  }
}

// ======================================================================
// host-side orchestration
// ======================================================================
extern "C" void kernel_launch(void* const* d_in, const int* in_sizes, int n_in,
                              void* d_out, int out_size, void* d_ws, size_t ws_size,
                              hipStream_t stream)
{
  (void)in_sizes; (void)n_in; (void)out_size; (void)ws_size;
  const int*   ids    = (const int*)d_in[0];
  const float* emb    = (const float*)d_in[1];
  const float* ln_w   = (const float*)d_in[2];
  const float* inw    = (const float*)d_in[3];
  const float* convw  = (const float*)d_in[4];
  const float* convb  = (const float*)d_in[5];
  const float* dtbias = (const float*)d_in[6];
  const float* alog   = (const float*)d_in[7];
  const float* dskip  = (const float*)d_in[8];
  const float* gw     = (const float*)d_in[9];
  const float* outw   = (const float*)d_in[10];
  const float* nfw    = (const float*)d_in[11];
  const float* lmw    = (const float*)d_in[12];
  float* out = (float*)d_out;

  float* p   = (float*)d_ws;
  float* res = p; p += (size_t)LSEQ * DMODEL;
  float* hb  = p; p += (size_t)LSEQ * DMODEL;
  float* nrm = p; p += (size_t)LSEQ * DMODEL;
  float* zx  = p; p += (size_t)LSEQ * DINPROJ;
  float* xbc = p; p += (size_t)LSEQ * CONVDIM;
  float* dtb = p; p += (size_t)LSEQ * NHEADS;
  float* yb  = p; p += (size_t)LSEQ * DINNER;
  float* gt  = p; p += (size_t)LSEQ * DINNER;
  float* st  = p; p += (size_t)NCHUNK * NHEADS * HEADDIM * DSTATE;
  float* pv  = p; p += (size_t)NCHUNK * NHEADS * HEADDIM * DSTATE;
  float* cd  = p; p += (size_t)NCHUNK * NHEADS * CHUNK;

  embed_gather<<<(LSEQ * DMODEL + 255) / 256, 256, 0, stream>>>(ids, emb, hb, res);

  for (int L = 0; L < NLAYERS; ++L) {
    rms_residual<<<LSEQ, 256, 0, stream>>>(hb, res, nrm, ln_w + L * DMODEL);
    gemm_tn_wmma<<<dim3((DINPROJ + 127) / 128, LSEQ / 128), 256, 0, stream>>>(
        nrm, inw + (size_t)L * DINPROJ * DMODEL, zx, LSEQ, DINPROJ, DMODEL);
    dt_softplus<<<(LSEQ * NHEADS + 255) / 256, 256, 0, stream>>>(
        zx, dtbias + L * NHEADS, dtb);
    conv_silu<<<(LSEQ * CONVDIM + 255) / 256, 256, 0, stream>>>(
        zx, convw + (size_t)L * CONVDIM * DCONV, convb + (size_t)L * CONVDIM, xbc);
    ssd_local<<<NCHUNK * NHEADS, 256, 0, stream>>>(
        xbc, dtb, alog + L * NHEADS, dskip + L * NHEADS, yb, st, cd);
    ssd_scan<<<NHEADS, 256, 0, stream>>>(st, cd, pv);
    ssd_cross<<<NCHUNK * NHEADS, 256, 0, stream>>>(xbc, pv, cd, yb);
    gate_rms<<<LSEQ, 256, 0, stream>>>(yb, zx, gw + L * DINNER, gt);
    gemm_tn_wmma<<<dim3(DMODEL / 128, LSEQ / 128), 256, 0, stream>>>(
        gt, outw + (size_t)L * DMODEL * DINNER, hb, LSEQ, DMODEL, DINNER);
  }

  rms_residual<<<LSEQ, 256, 0, stream>>>(hb, res, nrm, nfw);
  gemm_tn_wmma<<<dim3(VOCAB / 128, LSEQ / 128), 256, 0, stream>>>(
      nrm, lmw, out, LSEQ, VOCAB, DMODEL);
}